// GaussCopNormLoss_22823456211076
// MI455X (gfx1250) — compile-verified
//
#include <hip/hip_runtime.h>
#include <hip/hip_bf16.h>

typedef __attribute__((ext_vector_type(2))) float v2f;
typedef __attribute__((ext_vector_type(8))) float v8f;

#define LOSS_EPS 0.0001f

// ds_swizzle_b32 group-of-32 xor pattern: offset[14:10]=xor, [9:5]=or, [4:0]=and
#define SWZ_XOR(m) (((m) << 10) | 0x1f)

__device__ __forceinline__ float swz_add(float s, int imm_off) {
    // all-lanes-active xor butterfly step in a single DS op (no LDS memory touched)
    return s; // placeholder overwritten below by macro users
}

// ---------------------------------------------------------------------------
// Wave(32)-level sum reduction using V_WMMA_F32_16X16X4_F32.
// A = 16x4 all-ones (2 VGPRs/lane), B packs each lane's partial in component 0
// (4x16 = 64 f32 = 2 VGPRs/lane).  D[m][n] = sum_k B[k][n]  (column sums,
// identical for every row m since A's rows are identical).  D[0][0..15] lives
// in d[0] of lanes 0..15 (lanes 16..31 hold the identical M=8 row), so a
// 4-step xor butterfly over the 16 column sums leaves the full wave total in
// every lane.  EXEC must be all ones at the call site.
// ---------------------------------------------------------------------------
__device__ __forceinline__ float wave_reduce_wmma(float v) {
    v2f a; a[0] = 1.0f; a[1] = 1.0f;     // ones matrix
    v2f b; b[0] = v;    b[1] = 0.0f;     // lane partial
    v8f c = {};
    v8f d = __builtin_amdgcn_wmma_f32_16x16x4_f32(
        /*neg_a=*/false, a, /*neg_b=*/false, b,
        /*c_mod=*/(short)0, c, /*reuse_a=*/false, /*reuse_b=*/false);
    float s = d[0];
    s += __int_as_float(__builtin_amdgcn_ds_swizzle(__float_as_int(s), SWZ_XOR(1)));
    s += __int_as_float(__builtin_amdgcn_ds_swizzle(__float_as_int(s), SWZ_XOR(2)));
    s += __int_as_float(__builtin_amdgcn_ds_swizzle(__float_as_int(s), SWZ_XOR(4)));
    s += __int_as_float(__builtin_amdgcn_ds_swizzle(__float_as_int(s), SWZ_XOR(8)));
    return s;
}

__device__ __forceinline__ float clamp3(float x) {
    return fminf(fmaxf(x, -3.0f), 3.0f);
}

// ---------------------------------------------------------------------------
// Kernel 1: streaming pass, one partial sum (ce + copula) per block.
// ---------------------------------------------------------------------------
__global__ __launch_bounds__(256)
void gausscop_partial_kernel(const float* __restrict__ pred_r,
                             const float* __restrict__ pred_params,
                             const float* __restrict__ pred_corr,
                             const float* __restrict__ y_true,
                             float* __restrict__ partials,
                             int n) {
    const int tid    = blockIdx.x * blockDim.x + threadIdx.x;
    const int stride = gridDim.x * blockDim.x;

    const float2* __restrict__ pr  = (const float2*)pred_r;      // (B,2) 8B rows
    const float2* __restrict__ pp  = (const float2*)pred_params; // (B,6) as 3x float2
    const float4* __restrict__ yt  = (const float4*)y_true;      // (B,4) 16B rows

    float acc = 0.0f;
    for (int i = tid; i < n; i += stride) {
        // prefetch one grid-stride ahead into GL2 (locality 1 -> DEV scope;
        // dataset (120 MB) fits in the 192 MB L2, so replays hit L2)
        int ip = i + stride;
        if (ip < n) {
            __builtin_prefetch(&yt[ip], 0, 1);
            __builtin_prefetch(&pp[3 * (size_t)ip], 0, 1);
            __builtin_prefetch(&pred_corr[3 * (size_t)ip], 0, 1);
            __builtin_prefetch(&pr[ip], 0, 1);
        }

        float2 r  = pr[i];
        float2 pa = pp[3 * (size_t)i + 0];   // log(rtop), log(s_rtop)
        float2 pb = pp[3 * (size_t)i + 1];   // log(top),  log(s_top)
        float2 pc = pp[3 * (size_t)i + 2];   // log(close),log(s_close)
        float r12 = pred_corr[3 * (size_t)i + 0];
        float r13 = pred_corr[3 * (size_t)i + 1];
        float r23 = pred_corr[3 * (size_t)i + 2];
        float4 y  = yt[i];                   // revert, Rtop, Top, Close

        float rtop    = __expf(pa.x), s_rtop  = __expf(pa.y);
        float top     = __expf(pb.x), s_top   = __expf(pb.y);
        float close_  = __expf(pc.x), s_close = __expf(pc.y);

        float detR = 1.0f + 2.0f * r12 * r13 * r23
                   - r12 * r12 - r13 * r13 - r23 * r23;
        float inv_det = 1.0f / detR;
        float i00 = (1.0f - r23 * r23) * inv_det;
        float i01 = (r13 * r23 - r12) * inv_det;
        float i02 = (r12 * r23 - r13) * inv_det;
        float i11 = (1.0f - r13 * r13) * inv_det;
        float i12 = (r12 * r13 - r23) * inv_det;
        float i22 = (1.0f - r12 * r12) * inv_det;

        // z = clip(sqrt2*erfinv(2*clip(ndtr(t),eps,1-eps)-1), -3, 3) == clip(t,-3,3)
        float z1 = clamp3((y.y - rtop)   / s_rtop);
        float z2 = clamp3((y.z - top)    / s_top);
        float z3 = clamp3((y.w - close_) / s_close);

        float exp_term = 0.5f * ((i00 - 1.0f) * z1 * z1 +
                                 (i11 - 1.0f) * z2 * z2 +
                                 (i22 - 1.0f) * z3 * z3)
                       + (i01 * z1 * z2 + i02 * z1 * z3 + i12 * z2 * z3);
        float c_gauss = 0.5f * __logf(fmaxf(detR, LOSS_EPS)) + exp_term;

        // nll(u,s,x) = 0.5*((u-x)/max(s,eps))^2 + log(s);  log(s) == param
        float d1 = (rtop   - y.y) / fmaxf(s_rtop,  LOSS_EPS);
        float d2 = (top    - y.z) / fmaxf(s_top,   LOSS_EPS);
        float d3 = (close_ - y.w) / fmaxf(s_close, LOSS_EPS);
        float nll_sum = 0.5f * (d1 * d1 + d2 * d2 + d3 * d3) + pa.y + pb.y + pc.y;

        float copula = y.x * (c_gauss + nll_sum);

        // cross-entropy over 2 logits
        float mx  = fmaxf(r.x, r.y);
        float lse = mx + __logf(__expf(r.x - mx) + __expf(r.y - mx));
        float sel = ((int)y.x != 0) ? r.y : r.x;
        float ce  = lse - sel;

        acc += ce + copula;
    }

    // block reduction: WMMA wave sums -> LDS -> thread 0
    __shared__ float wsum[8];
    float ws = wave_reduce_wmma(acc);
    int lane = threadIdx.x & 31;
    int wid  = threadIdx.x >> 5;
    if (lane == 0) wsum[wid] = ws;
    __syncthreads();
    if (threadIdx.x == 0) {
        float t = 0.0f;
        #pragma unroll
        for (int w = 0; w < 8; ++w) t += wsum[w];
        partials[blockIdx.x] = t;
    }
}

// ---------------------------------------------------------------------------
// Kernel 2: single block reduces the block partials deterministically.
// ---------------------------------------------------------------------------
__global__ __launch_bounds__(256)
void gausscop_finalize_kernel(const float* __restrict__ partials, int np,
                              float* __restrict__ out, float inv_n) {
    float acc = 0.0f;
    for (int i = threadIdx.x; i < np; i += 256) acc += partials[i];

    __shared__ float wsum[8];
    float ws = wave_reduce_wmma(acc);
    int lane = threadIdx.x & 31;
    int wid  = threadIdx.x >> 5;
    if (lane == 0) wsum[wid] = ws;
    __syncthreads();
    if (threadIdx.x == 0) {
        float t = 0.0f;
        #pragma unroll
        for (int w = 0; w < 8; ++w) t += wsum[w];
        out[0] = t * inv_n;
    }
}

extern "C" void kernel_launch(void* const* d_in, const int* in_sizes, int n_in,
                              void* d_out, int out_size, void* d_ws, size_t ws_size,
                              hipStream_t stream) {
    const float* pred_r      = (const float*)d_in[0];   // (B,2)
    const float* pred_params = (const float*)d_in[1];   // (B,6)
    const float* pred_corr   = (const float*)d_in[2];   // (B,3)
    const float* y_true      = (const float*)d_in[3];   // (B,4)
    float* out = (float*)d_out;

    const int n = in_sizes[0] / 2;   // B

    int nblk = 1024;
    size_t ws_elems = ws_size / sizeof(float);
    if ((size_t)nblk > ws_elems) nblk = (int)ws_elems;
    if (nblk < 1) nblk = 1;

    float* partials = (float*)d_ws;

    gausscop_partial_kernel<<<nblk, 256, 0, stream>>>(
        pred_r, pred_params, pred_corr, y_true, partials, n);
    gausscop_finalize_kernel<<<1, 256, 0, stream>>>(
        partials, nblk, out, 1.0f / (float)n);
}